// spatial_new_11957188952741
// MI455X (gfx1250) — compile-verified
//
#include <hip/hip_runtime.h>

typedef __attribute__((ext_vector_type(2))) float v2f;
typedef __attribute__((ext_vector_type(4))) float v4f;
typedef __attribute__((ext_vector_type(8))) float v8f;

#define B_    32
#define C_    256
#define H_    56
#define W_    56
#define HW_   (H_ * W_)      // 3136
#define CHW_  (C_ * HW_)     // 802816
#define WB_   (W_ / 2)       // 28 blocks per row
#define NBLK_ (WB_ * WB_)    // 784 blocks per batch image
#define TILES_ (NBLK_ / 16)  // 49 tiles of 16 blocks

// -----------------------------------------------------------------------------
// Kernel 1: per-(batch, 16-block tile) masked box-sum via f32 WMMA.
//   A[m,k] = x[b,c,pos(m,k)] * mask[c,pos(m,k)]   (16 blocks x 4 window slots)
//   B      = all-ones 4x16  =>  D[m,n] = sum_k A[m,k], accumulated over C.
// 8 waves x 32 channels each, 512B LDS cross-wave reduce, gate -> d_ws.
// -----------------------------------------------------------------------------
__global__ __launch_bounds__(256) void block_sum_gate_kernel(
    const float* __restrict__ x, const float* __restrict__ mask,
    float* __restrict__ gate)
{
    __shared__ float partials[8][16];

    const int wg   = blockIdx.x;
    const int b    = wg / TILES_;
    const int tile = wg - b * TILES_;
    const int tid  = threadIdx.x;
    const int lane = tid & 31;
    const int wave = tid >> 5;

    // f32 WMMA A layout: lanes 0-15 carry (M=lane, K=0..1), lanes 16-31 carry
    // (M=lane-16, K=2..3).  K slots = the 4 pixels of the 2x2 window:
    // lane<16 -> top row of the block, lane>=16 -> bottom row.
    const int m  = lane & 15;            // block within tile (= WMMA row M)
    const int fb = tile * 16 + m;        // flat block index in 28x28 grid
    const int br = fb / WB_;
    const int bc = fb - br * WB_;
    const int dr = lane >> 4;            // 0: K={0,1}, 1: K={2,3}
    const int sp = (2 * br + dr) * W_ + 2 * bc;   // even -> 8B aligned float2

    const float* xp = x    + (size_t)b * CHW_ + sp;
    const float* mp = mask + sp;

    v8f acc = {};
    v2f ones; ones.x = 1.0f; ones.y = 1.0f;   // B matrix: all ones

    const int c0 = wave * 32;                 // 8 waves x 32 channels = 256
    #pragma unroll 4
    for (int i = 0; i < 32; ++i) {
        const size_t coff = (size_t)(c0 + i) * HW_;
        v2f xv = *(const v2f*)(xp + coff);
        v2f mv = *(const v2f*)(mp + coff);
        v2f a;
        a.x = xv.x * mv.x;
        a.y = xv.y * mv.y;
        acc = __builtin_amdgcn_wmma_f32_16x16x4_f32(
            /*neg_a=*/false, a, /*neg_b=*/false, ones,
            /*c_mod=*/(short)0, acc, /*reuse_a=*/false, /*reuse_b=*/false);
    }

    // D columns are identical; lane 0 holds sums for m=0..7 (VGPR j -> M=j),
    // lane 16 holds m=8..15.
    if (lane == 0) {
        #pragma unroll
        for (int j = 0; j < 8; ++j) partials[wave][j] = acc[j];
    }
    if (lane == 16) {
        #pragma unroll
        for (int j = 0; j < 8; ++j) partials[wave][8 + j] = acc[j];
    }
    __syncthreads();

    if (tid < 16) {
        float s = 0.0f;
        #pragma unroll
        for (int w = 0; w < 8; ++w) s += partials[w][tid];
        gate[b * NBLK_ + tile * 16 + tid] = (s > 0.0f) ? 1.0f : 0.0f;
    }
}

// -----------------------------------------------------------------------------
// Kernel 2: out = x * gate, float4-vectorized stream.  x re-read hits the
// 192MB L2 (whole tensor is 103MB), so HBM traffic stays ~read-once/write-once.
// 4 consecutive columns span exactly 2 spatial blocks (W=56 % 4 == 0).
// -----------------------------------------------------------------------------
__global__ __launch_bounds__(256) void apply_gate_kernel(
    const float* __restrict__ x, const float* __restrict__ gate,
    float* __restrict__ out)
{
    const size_t i4   = (size_t)blockIdx.x * 256 + threadIdx.x;
    const size_t base = i4 * 4;

    const int b  = (int)(base / CHW_);
    const int sp = (int)(base % HW_);
    const int h  = sp / W_;
    const int w  = sp - h * W_;

    const float* grow = gate + b * NBLK_ + (h >> 1) * WB_;
    const float g0 = grow[(w >> 1)];
    const float g1 = grow[(w >> 1) + 1];

    __builtin_prefetch(x + base + 16384, 0, 1);  // global_prefetch_b8 (speculative)

    v4f xv = *(const v4f*)(x + base);
    v4f o;
    o.x = xv.x * g0;
    o.y = xv.y * g0;
    o.z = xv.z * g1;
    o.w = xv.w * g1;
    *(v4f*)(out + base) = o;
}

extern "C" void kernel_launch(void* const* d_in, const int* in_sizes, int n_in,
                              void* d_out, int out_size, void* d_ws, size_t ws_size,
                              hipStream_t stream) {
    const float* x    = (const float*)d_in[0];
    const float* mask = (const float*)d_in[1];
    // d_in[2] = filt_size (== 2, hardcoded throughout)
    float* out  = (float*)d_out;
    float* gate = (float*)d_ws;          // needs 32*784*4 = 100,352 bytes

    block_sum_gate_kernel<<<dim3(B_ * TILES_), dim3(256), 0, stream>>>(x, mask, gate);

    const int n4blocks = (B_ * CHW_) / 4 / 256;   // 25088, exact
    apply_gate_kernel<<<dim3(n4blocks), dim3(256), 0, stream>>>(x, gate, out);
}